// DWTLayer_70334384439935
// MI455X (gfx1250) — compile-verified
//
#include <hip/hip_runtime.h>

typedef __attribute__((ext_vector_type(2))) float v2f;
typedef __attribute__((ext_vector_type(8))) float v8f;

// Haar 2x2 DWT, (8,512,512,16) f32 -> 4 x (8,256,256,16) f32 (LL,LH,HL,HH).
// Bandwidth-bound (256 MiB traffic, ~11.5us at 23.3 TB/s). Butterfly done with
// V_WMMA_F32_16X16X4_F32:
//   A (16x4): +-0.5 coefficients, rows 0-3 (block w2) / rows 8-11 (block w2+1)
//   B (4x16): K = {a,b,c,d}, N = 16 channels; lanes 0-15 read the even image
//             row, lanes 16-31 the odd row (matches the B VGPR layout).
// Two accumulating WMMAs fill D rows {0-3, 8-11}; D VGPR r = row r (lanes
// 0-15) | row r+8 (lanes 16-31), so acc[0..3] are each 128 contiguous output
// bytes per plane -> four full-width NT stores per block pair.
//
// Addressing: with p = b*32768 + h2*128 + w2p,
//   input  float offset = 128*p - 96*w2p,   output float offset = 32*p.
// nw (wave stride) is a multiple of 128, so w2p is loop-invariant and all
// bases advance by constant strides. readfirstlane keeps every base wave-
// uniform: saddr-form loads/stores, all per-iteration address math on SALU.
// Each output plane gets its own uniform base pointer so the per-lane store
// offset stays tiny (<= 124 B) and shared across the 4 stores.

__global__ __launch_bounds__(256) void dwt_haar_wmma(const float* __restrict__ x,
                                                     float* __restrict__ out) {
  const int lane = (int)(threadIdx.x & 31);
  const int half = lane >> 4;      // 0 = lanes 0-15, 1 = lanes 16-31
  const int ch   = lane & 15;

  // ---- constant A operands (16x4 f32) ----
  // A VGPR0 holds K = 2*half, A VGPR1 holds K = 2*half+1, M = lane&15
  auto coef = [](int t, int k) -> float {
    // t: 0=LL(++++), 1=LH(+-+-), 2=HL(++--), 3=HH(+--+); k: 0=a,1=b,2=c,3=d
    int neg = ((t & 1) & (k & 1)) ^ (((t >> 1) & 1) & ((k >> 1) & 1));
    return neg ? -0.5f : 0.5f;
  };
  const int m  = ch;
  const int k0 = 2 * half;
  const int k1 = 2 * half + 1;

  v2f a1, a2;   // coefficients targeting D rows 0-3 / rows 8-11
  a1[0] = (m < 4)            ? coef(m,     k0) : 0.0f;
  a1[1] = (m < 4)            ? coef(m,     k1) : 0.0f;
  a2[0] = (m >= 8 && m < 12) ? coef(m - 8, k0) : 0.0f;
  a2[1] = (m >= 8 && m < 12) ? coef(m - 8, k1) : 0.0f;

  // ---- geometry ----
  const int       ROW  = 512 * 16;              // input row stride (floats)
  const int       NOUT = 8 * 256 * 256 * 16;    // output plane size (floats)
  const long long totalPairs = 262144;          // 8*256*128 block pairs

  // wave id / wave count, forced wave-uniform (scalar) for saddr addressing
  const int wv = __builtin_amdgcn_readfirstlane(
      (int)((blockIdx.x * blockDim.x + threadIdx.x) >> 5));
  const int nw = __builtin_amdgcn_readfirstlane(
      (int)((gridDim.x * blockDim.x) >> 5));    // multiple of 128 by launch

  if (wv >= totalPairs) return;                 // wave-uniform guard

  const int w2p = wv & 127;                     // loop-invariant (nw % 128 == 0)

  // wave-uniform base pointers (one per output plane)
  const float* inPtr = x + 128LL * wv - 96LL * w2p;
  float* pLL = out + 32LL * wv;
  float* pLH = pLL + (long long)NOUT;
  float* pHL = pLL + 2LL * NOUT;
  float* pHH = pLL + 3LL * NOUT;
  const long long inStep  = 128LL * nw;         // floats per iteration
  const long long outStep = 32LL  * nw;

  // per-lane, loop-invariant small offsets
  const int inOfs  = half * ROW + ch;           // +0/+16/+32/+48 for the 4 taps
  const int outOfs = half * 16 + ch;            // shared by all 4 plane stores

  const int niter = (int)((totalPairs - wv + nw - 1) / nw);

  for (int it = 0; it < niter; ++it) {
    // B operands: block w2 at +0/+16 floats, block w2+1 at +32/+48 floats
    v2f b0, b1;
    b0[0] = __builtin_nontemporal_load(inPtr + inOfs + 0);
    b0[1] = __builtin_nontemporal_load(inPtr + inOfs + 16);
    b1[0] = __builtin_nontemporal_load(inPtr + inOfs + 32);
    b1[1] = __builtin_nontemporal_load(inPtr + inOfs + 48);

    v8f acc = {};
    acc = __builtin_amdgcn_wmma_f32_16x16x4_f32(false, a1, false, b0,
                                                (short)0, acc, false, false);
    acc = __builtin_amdgcn_wmma_f32_16x16x4_f32(false, a2, false, b1,
                                                (short)0, acc, false, false);

    __builtin_nontemporal_store(acc[0], pLL + outOfs);
    __builtin_nontemporal_store(acc[1], pLH + outOfs);
    __builtin_nontemporal_store(acc[2], pHL + outOfs);
    __builtin_nontemporal_store(acc[3], pHH + outOfs);

    inPtr += inStep;
    pLL += outStep;
    pLH += outStep;
    pHL += outStep;
    pHH += outStep;
  }
}

extern "C" void kernel_launch(void* const* d_in, const int* in_sizes, int n_in,
                              void* d_out, int out_size, void* d_ws, size_t ws_size,
                              hipStream_t stream) {
  (void)in_sizes; (void)n_in; (void)out_size; (void)d_ws; (void)ws_size;
  const float* x = (const float*)d_in[0];
  float* out = (float*)d_out;
  // 2048 WGs x 256 threads = 16384 waves (nw multiple of 128);
  // each wave streams 16 block pairs at constant stride.
  dwt_haar_wmma<<<dim3(2048), dim3(256), 0, stream>>>(x, out);
}